// RWKVRegionCell_6803228197163
// MI455X (gfx1250) — compile-verified
//
#include <hip/hip_runtime.h>
#include <hip/hip_bf16.h>

typedef __attribute__((ext_vector_type(2))) float v2f;
typedef __attribute__((ext_vector_type(8))) float v8f;

#define DDIM 8192
#define KSPLIT 2048   // K-range per block (4 splits over 8192)
#define KCH 64        // K chunk staged in LDS per iteration
#define LDW 68        // LDS row pitch in floats (bank-conflict-free fragments)
#define KWTA_K 1024   // D/8

// ---------------------------------------------------------------------------
// Phase 1: x = rmsnorm(x_in + pred) * norm_w ; zero the 3 matvec accumulators
// ---------------------------------------------------------------------------
__global__ void __launch_bounds__(1024) rms_kernel(
    const float* __restrict__ x_in, const float* __restrict__ pred,
    const float* __restrict__ norm_w, float* __restrict__ x,
    float* __restrict__ acc3 /* 3*8192 floats to zero */)
{
    __shared__ float red[1024];
    __shared__ float sscale;
    const int tid = threadIdx.x;
    float vals[8];
    float s = 0.f;
#pragma unroll
    for (int j = 0; j < 8; ++j) {
        int i = tid + j * 1024;
        float t = x_in[i] + pred[i];
        vals[j] = t;
        s += t * t;
    }
    red[tid] = s;
    __syncthreads();
    for (int off = 512; off > 0; off >>= 1) {
        if (tid < off) red[tid] += red[tid + off];
        __syncthreads();
    }
    if (tid == 0) sscale = rsqrtf(red[0] * (1.0f / (float)DDIM) + 1e-6f);
    __syncthreads();
    const float sc = sscale;
#pragma unroll
    for (int j = 0; j < 8; ++j) {
        int i = tid + j * 1024;
        x[i] = vals[j] * sc * norm_w[i];
    }
#pragma unroll
    for (int j = 0; j < 24; ++j)
        acc3[tid + j * 1024] = 0.f;
}

// ---------------------------------------------------------------------------
// WMMA fp32 GEMV body: one wave computes a 16-row tile over a 2048-wide
// K-split using V_WMMA_F32_16X16X4_F32, then atomically accumulates.
// Two independent accumulators break the WMMA->WMMA RAW chain.
// ---------------------------------------------------------------------------
__device__ __forceinline__ void gemv_wmma_body(
    const float* __restrict__ W,   // row-major [DDIM x DDIM]
    const float* __restrict__ xv,  // [DDIM]
    float* __restrict__ acc,       // [DDIM], atomic accumulate target
    int tile64, int ksp,
    float* xs, float (*wt)[16 * LDW])
{
    const int tid  = threadIdx.x;
    const int lane = tid & 31;
    const int wave = tid >> 5;
    const int kbase = ksp * KSPLIT;

    // Stage this block's x slice (shared by all 4 waves)
    for (int j = tid; j < KSPLIT; j += 128) xs[j] = xv[kbase + j];
    __syncthreads();

    const int rowbase = tile64 * 64 + wave * 16;
    const float* Wrow = W + (size_t)rowbase * DDIM + kbase;
    float* wl = wt[wave];

    const int hi2 = (lane >> 4) << 1;  // 0 for lanes 0-15, 2 for lanes 16-31
    const int rm  = lane & 15;

    v8f c0 = {0.f, 0.f, 0.f, 0.f, 0.f, 0.f, 0.f, 0.f};
    v8f c1 = {0.f, 0.f, 0.f, 0.f, 0.f, 0.f, 0.f, 0.f};

    for (int kc = 0; kc < KSPLIT; kc += KCH) {
        // Coalesced stage of W[16 x 64] chunk into per-wave LDS (pitch LDW).
        // 256 float4 words; lanes 0-15 cover one 256B row segment each pass.
#pragma unroll
        for (int i = 0; i < 8; ++i) {
            int w  = i * 32 + lane;
            int rr = w >> 4;
            int c4 = w & 15;
            const float4 d = *(const float4*)(Wrow + (size_t)rr * DDIM + kc + c4 * 4);
            *(float4*)(&wl[rr * LDW + c4 * 4]) = d;
        }
        // Prefetch next chunk (global_prefetch_b8)
        if (kc + KCH < KSPLIT)
            __builtin_prefetch(Wrow + (size_t)rm * DDIM + kc + KCH, 0, 0);

        // 16 WMMA steps of K=4 each; even/odd steps go to independent
        // accumulators so consecutive WMMAs have no RAW dependency.
#pragma unroll
        for (int kk = 0; kk < KCH; kk += 8) {
            v2f a0, b0, a1, b1;
            a0.x = wl[rm * LDW + kk + hi2];
            a0.y = wl[rm * LDW + kk + hi2 + 1];
            b0.x = xs[kc + kk + hi2];          // x broadcast into all 16 cols
            b0.y = xs[kc + kk + hi2 + 1];
            a1.x = wl[rm * LDW + kk + 4 + hi2];
            a1.y = wl[rm * LDW + kk + 4 + hi2 + 1];
            b1.x = xs[kc + kk + 4 + hi2];
            b1.y = xs[kc + kk + 4 + hi2 + 1];
            c0 = __builtin_amdgcn_wmma_f32_16x16x4_f32(
                     false, a0, false, b0, (short)0, c0, false, false);
            c1 = __builtin_amdgcn_wmma_f32_16x16x4_f32(
                     false, a1, false, b1, (short)0, c1, false, false);
        }
    }

    const v8f c = c0 + c1;

    // C layout: VGPR j -> (lane 0: row j), (lane 16: row j+8)
    if (lane == 0) {
#pragma unroll
        for (int j = 0; j < 8; ++j) atomicAdd(&acc[rowbase + j], c[j]);
    } else if (lane == 16) {
#pragma unroll
        for (int j = 0; j < 8; ++j) atomicAdd(&acc[rowbase + 8 + j], c[j]);
    }
}

// ---------------------------------------------------------------------------
// Phase 2: fused r/k/v matvecs. grid = 3 matrices * 128 row-tiles * 4 K-splits
// ---------------------------------------------------------------------------
__global__ void __launch_bounds__(128) gemv3_kernel(
    const float* __restrict__ Wr, const float* __restrict__ Wk,
    const float* __restrict__ Wv, const float* __restrict__ x,
    float* __restrict__ accR, float* __restrict__ accK, float* __restrict__ accV)
{
    __shared__ float xs[KSPLIT];
    __shared__ float wt[4][16 * LDW];
    const int b   = blockIdx.x;       // 0..1535
    const int m   = b >> 9;           // 0..2
    const int rem = b & 511;
    const int t   = rem >> 2;         // row tile 0..127
    const int s   = rem & 3;          // K split 0..3
    const float* W = (m == 0) ? Wr : (m == 1) ? Wk : Wv;
    float* acc     = (m == 0) ? accR : (m == 1) ? accK : accV;
    gemv_wmma_body(W, x, acc, t, s, xs, wt);
}

// ---------------------------------------------------------------------------
// Phase 3: elementwise gate/RoPE/state update -> y ; also init h = x
// ---------------------------------------------------------------------------
__global__ void __launch_bounds__(256) elem_kernel(
    const float* __restrict__ accR, const float* __restrict__ accK,
    const float* __restrict__ accV, const float* __restrict__ state_num,
    const float* __restrict__ state_den, const float* __restrict__ decay,
    const float* __restrict__ W_time, const float* __restrict__ step_pos,
    const int* __restrict__ dt, const float* __restrict__ x,
    float* __restrict__ y, float* __restrict__ h)
{
    const int i = blockIdx.x * blockDim.x + threadIdx.x;  // 0..8191
    const float r = 1.f / (1.f + expf(-accR[i]));
    float v = accV[i];
    if (i < 32) {  // RoPE over first 16 pairs
        int p = i >> 1;
        float th = step_pos[0] * W_time[p];
        float cth = cosf(th), sth = sinf(th);
        float v0 = accV[2 * p], v1 = accV[2 * p + 1];
        v = (i & 1) ? (v0 * sth + v1 * cth) : (v0 * cth - v1 * sth);
    }
    const float w   = expf(accK[i]);
    const float sg  = 1.f / (1.f + expf(-decay[i]));
    const float lam = sg * sg;
    const float mul = powf(lam, (float)dt[0]) * lam;  // lam^dt * lam
    const float sn  = state_num[i] * mul + w * v;
    const float sd  = state_den[i] * mul + w;
    y[i] = r * (sn / (sd + 1e-9f));
    h[i] = x[i];  // h accumulates x + Wo@y in phase 4
}

// ---------------------------------------------------------------------------
// Phase 4: h += Wo @ y. grid = 128 row-tiles * 4 K-splits
// ---------------------------------------------------------------------------
__global__ void __launch_bounds__(128) gemv1_kernel(
    const float* __restrict__ Wo, const float* __restrict__ y,
    float* __restrict__ h)
{
    __shared__ float xs[KSPLIT];
    __shared__ float wt[4][16 * LDW];
    const int b = blockIdx.x;  // 0..511
    gemv_wmma_body(Wo, y, h, b >> 2, b & 3, xs, wt);
}

// ---------------------------------------------------------------------------
// Phase 5: exact kWTA. Bitwise binary search for the 1024-th largest |h|
// (positive-float bit patterns order like uints), then threshold.
// ---------------------------------------------------------------------------
__global__ void __launch_bounds__(1024) kwta_kernel(
    const float* __restrict__ h, float* __restrict__ out)
{
    __shared__ int cnt[1024];
    const int tid = threadIdx.x;
    float hv[8];
    unsigned u[8];
#pragma unroll
    for (int j = 0; j < 8; ++j) {
        int i = tid + j * 1024;
        hv[j] = h[i];
        u[j]  = __float_as_uint(hv[j]) & 0x7fffffffu;
    }
    unsigned cur = 0u;
    for (int bit = 30; bit >= 0; --bit) {
        unsigned cand = cur | (1u << bit);
        int c = 0;
#pragma unroll
        for (int j = 0; j < 8; ++j) c += (u[j] >= cand) ? 1 : 0;
        cnt[tid] = c;
        __syncthreads();
        for (int off = 512; off > 0; off >>= 1) {
            if (tid < off) cnt[tid] += cnt[tid + off];
            __syncthreads();
        }
        int tot = cnt[0];
        __syncthreads();
        if (tot >= KWTA_K) cur = cand;  // uniform across block
    }
#pragma unroll
    for (int j = 0; j < 8; ++j) {
        int i = tid + j * 1024;
        out[i] = (u[j] >= cur) ? hv[j] : 0.0f;
    }
}

// ---------------------------------------------------------------------------
extern "C" void kernel_launch(void* const* d_in, const int* in_sizes, int n_in,
                              void* d_out, int out_size, void* d_ws, size_t ws_size,
                              hipStream_t stream)
{
    const float* x_in      = (const float*)d_in[0];
    const float* state_num = (const float*)d_in[1];
    const float* state_den = (const float*)d_in[2];
    const float* pred      = (const float*)d_in[3];
    const float* norm_w    = (const float*)d_in[4];
    const float* Wr        = (const float*)d_in[5];
    const float* Wk        = (const float*)d_in[6];
    const float* Wv        = (const float*)d_in[7];
    const float* Wo        = (const float*)d_in[8];
    const float* decay     = (const float*)d_in[9];
    const float* W_time    = (const float*)d_in[10];
    const float* step_pos  = (const float*)d_in[11];
    const int*   dt        = (const int*)d_in[12];
    float* out = (float*)d_out;

    float* ws   = (float*)d_ws;
    float* wx   = ws;              // 8192 : normalized x
    float* accR = ws + 8192;       // 8192 : Wr@x
    float* accK = ws + 16384;      // 8192 : Wk@x
    float* accV = ws + 24576;      // 8192 : Wv@x
    float* wy   = ws + 32768;      // 8192 : y
    float* wh   = ws + 40960;      // 8192 : h = x + Wo@y

    rms_kernel<<<1, 1024, 0, stream>>>(x_in, pred, norm_w, wx, accR);
    gemv3_kernel<<<1536, 128, 0, stream>>>(Wr, Wk, Wv, wx, accR, accK, accV);
    elem_kernel<<<32, 256, 0, stream>>>(accR, accK, accV, state_num, state_den,
                                        decay, W_time, step_pos, dt, wx, wy, wh);
    gemv1_kernel<<<512, 128, 0, stream>>>(Wo, wy, wh);
    kwta_kernel<<<1, 1024, 0, stream>>>(wh, out);
}